// CCN_16303695855752
// MI455X (gfx1250) — compile-verified
//
#include <hip/hip_runtime.h>

#define B_ 8
#define N_ 2048
#define E_ 128
#define K_ 10

typedef float v2f __attribute__((ext_vector_type(2)));
typedef float v8f __attribute__((ext_vector_type(8)));

// ---------------------------------------------------------------------------
// kNN: per (b,n) find 10 smallest distances (self included), stable ties.
// One thread per node; batch coords staged in LDS (2048*2*4 = 16KB).
// ---------------------------------------------------------------------------
__global__ void __launch_bounds__(256)
knn_kernel(const float* __restrict__ loc, int* __restrict__ nbr) {
    __shared__ float sx[N_];
    __shared__ float sy[N_];
    const int b = blockIdx.x >> 3;      // 8 chunks of 256 rows per batch
    const int chunk = blockIdx.x & 7;
    const int tid = threadIdx.x;
    for (int i = tid; i < N_; i += 256) {
        sx[i] = loc[(b * N_ + i) * 2 + 0];
        sy[i] = loc[(b * N_ + i) * 2 + 1];
    }
    __syncthreads();

    const int n = chunk * 256 + tid;
    const float px = sx[n], py = sy[n];

    float bd[K_];
    int   bi[K_];
#pragma unroll
    for (int p = 0; p < K_; ++p) { bd[p] = 3.4e38f; bi[p] = 0; }

    for (int j = 0; j < N_; ++j) {
        const float dx = sx[j] - px;
        const float dy = sy[j] - py;
        const float d = dx * dx + dy * dy;   // sqrt is monotone: same top-k
        if (d < bd[K_ - 1]) {
            float cd = d; int ci = j;
#pragma unroll
            for (int p = 0; p < K_; ++p) {
                const bool lt = cd < bd[p];  // strict: stable (low index wins ties)
                const float td = bd[p]; const int ti = bi[p];
                bd[p] = lt ? cd : bd[p];
                bi[p] = lt ? ci : bi[p];
                cd = lt ? td : cd;
                ci = lt ? ti : ci;
            }
        }
    }
#pragma unroll
    for (int p = 0; p < K_; ++p) nbr[(b * N_ + n) * K_ + p] = bi[p];
}

// ---------------------------------------------------------------------------
// GEMM1 (fused): per wave, build A[16x128] = [F0 row; 10 nbd rows; 5 zero rows]
// on the fly, multiply by W_t1 (LDS-staged) with v_wmma_f32_16x16x4_f32,
// epilogue = +bias, leaky_relu(0.01), sum rows 0..10 -> F1[b,n,128].
// ---------------------------------------------------------------------------
__global__ void __launch_bounds__(256)
gemm1_kernel(const float* __restrict__ loc,
             const int*   __restrict__ nbr,
             const float* __restrict__ W_init, const float* __restrict__ b_init,
             const float* __restrict__ W_nb,   const float* __restrict__ b_nb,
             const float* __restrict__ W_t1,   const float* __restrict__ b_t1,
             float* __restrict__ F1) {
    __shared__ float s_w[E_ * E_];      // 64 KB of the 320 KB WGP LDS
    {
        const float4* src = (const float4*)W_t1;
        float4* dst = (float4*)s_w;
        for (int i = threadIdx.x; i < E_ * E_ / 4; i += 256) dst[i] = src[i];
    }
    __syncthreads();

    const int lane  = threadIdx.x & 31;
    const int wave  = threadIdx.x >> 5;
    const int idx   = blockIdx.x * 8 + wave;   // one (b,n) per wave
    const int b     = idx >> 11;               // / 2048
    const int n     = idx & (N_ - 1);
    const int row   = lane & 15;               // A-matrix M for this lane
    const int khalf = lane >> 4;               // 0: K%4 in {0,1}; 1: {2,3}
    const int col   = lane & 15;               // C-matrix N for this lane

    const float lx = loc[(b * N_ + n) * 2 + 0];
    const float ly = loc[(b * N_ + n) * 2 + 1];

    // Per-lane row generator: row 0 -> F0, rows 1..10 -> nbd, rows 11..15 -> 0
    float cx = 0.f, cy = 0.f;
    const float* w0 = W_init;
    const float* w1 = W_init + E_;
    const float* bb = b_init;
    const bool zero_row = (row > K_);
    if (row == 0) {
        cx = lx; cy = ly;
    } else if (row <= K_) {
        const int j = nbr[(b * N_ + n) * K_ + (row - 1)];
        cx = loc[j * 2 + 0] - lx;   // batch-0 coords (reference quirk)
        cy = loc[j * 2 + 1] - ly;
        w0 = W_nb; w1 = W_nb + E_; bb = b_nb;
    }

    // A tile in registers: lane holds its row's values at K = 4*kk + 2*khalf (+1)
    float a[64];
#pragma unroll
    for (int kk = 0; kk < 32; ++kk) {
        const int k0 = kk * 4 + khalf * 2;
        a[2 * kk + 0] = zero_row ? 0.f : (cx * w0[k0]     + cy * w1[k0]     + bb[k0]);
        a[2 * kk + 1] = zero_row ? 0.f : (cx * w0[k0 + 1] + cy * w1[k0 + 1] + bb[k0 + 1]);
    }

    float* out = F1 + (b * N_ + n) * E_;
    for (int nt = 0; nt < 8; ++nt) {
        v8f acc = {0.f, 0.f, 0.f, 0.f, 0.f, 0.f, 0.f, 0.f};
        const int c = nt * 16 + col;
#pragma unroll
        for (int kk = 0; kk < 32; ++kk) {
            const int k0 = kk * 4 + khalf * 2;
            v2f av; av[0] = a[2 * kk]; av[1] = a[2 * kk + 1];
            v2f bv; bv[0] = s_w[k0 * E_ + c]; bv[1] = s_w[(k0 + 1) * E_ + c];
            acc = __builtin_amdgcn_wmma_f32_16x16x4_f32(
                false, av, false, bv, (short)0, acc, false, false);
        }
        const float bias = b_t1[c];
        float partial = 0.f;
#pragma unroll
        for (int v = 0; v < 8; ++v) {
            const int M = v + khalf * 8;           // C layout: VGPR v -> row v / v+8
            float x = acc[v] + bias;
            x = x > 0.f ? x : 0.01f * x;           // leaky_relu
            partial += (M <= K_) ? x : 0.f;        // sum rows 0..10 only
        }
        const float total = partial + __shfl_xor(partial, 16, 32);
        if (khalf == 0) out[c] = total;
    }
}

// ---------------------------------------------------------------------------
// GEMM2 (fused): A rows = F1[0][nbr[b,n,0..9]], 6 zero rows; x W_t2;
// epilogue = +bias, leaky_relu, sum rows 0..9 -> h[b, 1+n, :].
// ---------------------------------------------------------------------------
__global__ void __launch_bounds__(256)
gemm2_kernel(const int*   __restrict__ nbr,
             const float* __restrict__ F1,
             const float* __restrict__ W_t2, const float* __restrict__ b_t2,
             float* __restrict__ h) {
    __shared__ float s_w[E_ * E_];
    {
        const float4* src = (const float4*)W_t2;
        float4* dst = (float4*)s_w;
        for (int i = threadIdx.x; i < E_ * E_ / 4; i += 256) dst[i] = src[i];
    }
    __syncthreads();

    const int lane  = threadIdx.x & 31;
    const int wave  = threadIdx.x >> 5;
    const int idx   = blockIdx.x * 8 + wave;
    const int b     = idx >> 11;
    const int n     = idx & (N_ - 1);
    const int row   = lane & 15;
    const int khalf = lane >> 4;
    const int col   = lane & 15;

    float a[64];
    if (row < K_) {
        const int j = nbr[(b * N_ + n) * K_ + row];
        const float* fr = F1 + j * E_;             // F1 of batch 0 (quirk)
#pragma unroll
        for (int kk = 0; kk < 32; ++kk) {
            const int k0 = kk * 4 + khalf * 2;
            a[2 * kk + 0] = fr[k0];
            a[2 * kk + 1] = fr[k0 + 1];
        }
    } else {
#pragma unroll
        for (int i = 0; i < 64; ++i) a[i] = 0.f;
    }

    float* out = h + (b * (N_ + 1) + 1 + n) * E_;
    for (int nt = 0; nt < 8; ++nt) {
        v8f acc = {0.f, 0.f, 0.f, 0.f, 0.f, 0.f, 0.f, 0.f};
        const int c = nt * 16 + col;
#pragma unroll
        for (int kk = 0; kk < 32; ++kk) {
            const int k0 = kk * 4 + khalf * 2;
            v2f av; av[0] = a[2 * kk]; av[1] = a[2 * kk + 1];
            v2f bv; bv[0] = s_w[k0 * E_ + c]; bv[1] = s_w[(k0 + 1) * E_ + c];
            acc = __builtin_amdgcn_wmma_f32_16x16x4_f32(
                false, av, false, bv, (short)0, acc, false, false);
        }
        const float bias = b_t2[c];
        float partial = 0.f;
#pragma unroll
        for (int v = 0; v < 8; ++v) {
            const int M = v + khalf * 8;
            float x = acc[v] + bias;
            x = x > 0.f ? x : 0.01f * x;
            partial += (M < K_) ? x : 0.f;         // sum rows 0..9
        }
        const float total = partial + __shfl_xor(partial, 16, 32);
        if (khalf == 0) out[c] = total;
    }
}

// ---------------------------------------------------------------------------
// h[b,0,:] = depot[b] @ W_dep + b_dep
// ---------------------------------------------------------------------------
__global__ void depot_kernel(const float* __restrict__ depot,
                             const float* __restrict__ W_dep,
                             const float* __restrict__ b_dep,
                             float* __restrict__ h) {
    const int t = blockIdx.x * blockDim.x + threadIdx.x;   // 1024 = B*E
    const int b = t >> 7;
    const int e = t & 127;
    h[b * (N_ + 1) * E_ + e] =
        depot[b * 2 + 0] * W_dep[e] + depot[b * 2 + 1] * W_dep[E_ + e] + b_dep[e];
}

// ---------------------------------------------------------------------------
// mean over axis 1 (2049 rows)
// ---------------------------------------------------------------------------
__global__ void mean_kernel(const float* __restrict__ h, float* __restrict__ m) {
    const int t = blockIdx.x * blockDim.x + threadIdx.x;   // 1024 = B*E
    const int b = t >> 7;
    const int e = t & 127;
    float s = 0.f;
    for (int r = 0; r < N_ + 1; ++r) s += h[(b * (N_ + 1) + r) * E_ + e];
    m[b * E_ + e] = s * (1.0f / (float)(N_ + 1));
}

// ---------------------------------------------------------------------------
extern "C" void kernel_launch(void* const* d_in, const int* in_sizes, int n_in,
                              void* d_out, int out_size, void* d_ws, size_t ws_size,
                              hipStream_t stream) {
    const float* loc    = (const float*)d_in[0];
    const float* depot  = (const float*)d_in[1];
    const float* W_init = (const float*)d_in[2];
    const float* b_init = (const float*)d_in[3];
    const float* W_nb   = (const float*)d_in[4];
    const float* b_nb   = (const float*)d_in[5];
    const float* W_t1   = (const float*)d_in[6];
    const float* b_t1   = (const float*)d_in[7];
    const float* W_t2   = (const float*)d_in[8];
    const float* b_t2   = (const float*)d_in[9];
    const float* W_dep  = (const float*)d_in[10];
    const float* b_dep  = (const float*)d_in[11];

    float* h        = (float*)d_out;                  // [8][2049][128]
    float* out_mean = h + B_ * (N_ + 1) * E_;         // [8][128]

    int*   nbr = (int*)d_ws;                                  // 640 KB
    float* F1  = (float*)((char*)d_ws + (1u << 20));          // 8 MB

    knn_kernel  <<<B_ * 8,      256, 0, stream>>>(loc, nbr);
    gemm1_kernel<<<B_ * N_ / 8, 256, 0, stream>>>(loc, nbr, W_init, b_init,
                                                  W_nb, b_nb, W_t1, b_t1, F1);
    gemm2_kernel<<<B_ * N_ / 8, 256, 0, stream>>>(nbr, F1, W_t2, b_t2, h);
    depot_kernel<<<4, 256, 0, stream>>>(depot, W_dep, b_dep, h);
    mean_kernel <<<4, 256, 0, stream>>>(h, out_mean);
}